// Decoder_81080392614208
// MI455X (gfx1250) — compile-verified
//
#include <hip/hip_runtime.h>
#include <hip/hip_bf16.h>

// ---------------------------------------------------------------------------
// Decoder: bit-unpack 23x22-bit codes, signed gather from fp16 table,
// scatter into rows 19:67 of a 0.5-filled [B,2,126,128] f32 tensor.
//
// Roofline: 1.06 GB streamed stores + ~0.2 GB table gathers @ 23.3 TB/s
// -> purely memory bound, no matmul -> no WMMA. CDNA5 levers used:
//   * global_load_async_to_lds_b64 (ASYNCcnt) to stage each batch's 2 KB
//     bit-row into LDS with one coalesced async copy per wave.
//   * NT stores (th:TH_STORE_NT) so the 1.06 GB output stream does not
//     evict the 201 MB table from the 192 MB L2.
//   * global_prefetch_b8 to warm each batch's 23 gather blocks.
//   * hot loop = b64 gather + fp16 math + NT b128 store, coalesced per wave.
// ---------------------------------------------------------------------------

typedef _Float16 hv4 __attribute__((ext_vector_type(4)));
typedef float    fv4 __attribute__((ext_vector_type(4)));

#define BATCH      8192
#define ROW_INTS   512          // 6 + 23*22
#define NCODES     23
#define NBITS      22
#define TABLE_LEN  131072       // 2^17
#define IDX_MASK   (TABLE_LEN - 1)
#define ROW_LO     19
#define ROW_HI     67
// per-batch output: [2][126][128] f32 = 32256 floats = 8064 float4
#define OUT_F4_PER_BATCH 8064

__global__ __launch_bounds__(256)
void decoder_kernel(const int* __restrict__ x,
                    const _Float16* __restrict__ table,
                    float* __restrict__ out)
{
    const int b   = blockIdx.x;
    const int tid = threadIdx.x;

    __shared__ int       s_bits[ROW_INTS];   // 2048 B staged bit-row
    __shared__ int       s_idx[NCODES];
    __shared__ _Float16  s_sgn[NCODES];

    // ---- stage 0: async-copy this batch's 2 KB bit-row into LDS -----------
    // 256 lanes x 8 B = 2048 B, one global_load_async_to_lds_b64 per lane.
    {
        const unsigned long long gbase =
            (unsigned long long)(const void*)(x + (size_t)b * ROW_INTS);
        const unsigned voff = (unsigned)tid * 8u;               // global byte off
        const unsigned lds  =                                   // LDS byte off
            (unsigned)(unsigned long long)(const void*)&s_bits[0] + (unsigned)tid * 8u;
        asm volatile("global_load_async_to_lds_b64 %0, %1, %2"
                     :: "v"(lds), "v"(voff), "s"(gbase) : "memory");
        asm volatile("s_wait_asynccnt 0" ::: "memory");
    }
    __syncthreads();   // all waves' async chunks now visible in LDS

    // ---- stage 1: decode 23 codes (one per thread), prefetch gather bases --
    if (tid < NCODES) {
        const int* pb = &s_bits[6 + tid * NBITS];
        int code = 0;
        #pragma unroll
        for (int a = 0; a < NBITS; ++a) {
            code |= (pb[a] & 1) << a;           // bits are {0,1} int32, LSB first
        }
        const int idx = code & IDX_MASK;        // codes % 131072
        s_idx[tid] = idx;
        s_sgn[tid] = (code > TABLE_LEN) ? (_Float16)(-1.0f) : (_Float16)(1.0f);

        // Warm L2/L0 for this code's [2,48,8] fp16 block (1536 B) before the
        // streaming store loop consumes it (gfx1250 global_prefetch_b8 path).
        const char* base = (const char*)(table + (size_t)idx * 768);
        #pragma unroll
        for (int off = 0; off < 1536; off += 128) {
            __builtin_prefetch(base + off, 0, 1);
        }
    }

    // ---- per-thread loop invariants (n4 == tid & 31 is constant) -----------
    const int n4   = tid & 31;                  // column group of 4 (n = 4*n4)
    const int wv   = tid >> 5;                  // wave id 0..7 -> row stride 8
    const int slot = n4 >> 1;                   // 0..15
    const int chb  = (n4 & 1) << 2;             // 0 or 4
    // columns 0..3 of slots 0..6 -> codes 0..6 (table ch 0..3)
    // columns 4..7 of slots 0..6 -> codes 7..13 (table ch 4..7)
    // slots 7..15 (all 8 ch)     -> codes 14..22
    const int code = (slot < 7) ? (chb ? slot + 7 : slot) : slot + 7;

    __syncthreads();

    const int       idx = s_idx[code];
    const _Float16  s   = s_sgn[code];
    const _Float16  h05 = (_Float16)0.5f;
    // table flat: idx*768 + h*384 + rr*8 + chb  (fp16 elements)
    const _Float16* __restrict__ gbase = table + (size_t)idx * 768 + chb;

    fv4* __restrict__ out4 =
        reinterpret_cast<fv4*>(out) + (size_t)b * OUT_F4_PER_BATCH + n4;

    const fv4 chalf = {0.5f, 0.5f, 0.5f, 0.5f};

    #pragma unroll
    for (int h = 0; h < 2; ++h) {
        const _Float16* __restrict__ gh   = gbase + h * 384;
        fv4* __restrict__            outh = out4 + h * (126 * 32);
        for (int r = wv; r < 126; r += 8) {      // wave-uniform r
            fv4 v;
            if (r >= ROW_LO && r < ROW_HI) {     // wave-uniform branch
                const hv4 g = *reinterpret_cast<const hv4*>(gh + (r - ROW_LO) * 8);
                // exact fp16 sequence of the reference: 0.5 + s*(g - 0.5)
                // (s*t is exact for s = +-1, so contraction is rounding-neutral)
                const hv4 val = h05 + s * (g - h05);
                v.x = (float)val.x;
                v.y = (float)val.y;
                v.z = (float)val.z;
                v.w = (float)val.w;
            } else {
                v = chalf;
            }
            // NT store: keep the 1.06 GB output stream out of L2 so the
            // 201 MB table stays resident for the gathers.
            __builtin_nontemporal_store(v, outh + r * 32);
        }
    }
}

extern "C" void kernel_launch(void* const* d_in, const int* in_sizes, int n_in,
                              void* d_out, int out_size, void* d_ws, size_t ws_size,
                              hipStream_t stream)
{
    (void)in_sizes; (void)n_in; (void)out_size; (void)d_ws; (void)ws_size;

    const int*      x     = (const int*)d_in[0];        // [8192, 512] int32
    const _Float16* table = (const _Float16*)d_in[1];   // [131072, 2, 48, 8] fp16
    float*          out   = (float*)d_out;              // [8192, 2, 126, 128] f32

    decoder_kernel<<<BATCH, 256, 0, stream>>>(x, table, out);
}